// ConnectionistAlignmentLoss_51367808860406
// MI455X (gfx1250) — compile-verified
//
#include <hip/hip_runtime.h>

// Problem constants (from the reference)
#define NEGV     (-1e30f)
#define BATCH    32
#define TLEN     1000
#define VOCAB    256
#define BLOCK    256            // 8 wave32 waves
#define COLS     4              // columns (target positions) per thread
#define NWAVES   (BLOCK / 32)

// ---------------------------------------------------------------------------
// CDNA5 async global->LDS copy (tracked with ASYNCcnt).
// GVS addressing: mem = SADDR(64b sgpr pair) + VADDR(32b vgpr offset).
// VDST vgpr carries the LDS byte address.
// ---------------------------------------------------------------------------
__device__ __forceinline__ unsigned lds_off(const void* p) {
    // Low 32 bits of a flat pointer into LDS == wave-relative LDS byte offset.
    return (unsigned)(unsigned long long)p;
}

__device__ __forceinline__ void async_load_b32(const float* gbase,
                                               unsigned goff_bytes,
                                               unsigned lds_addr) {
    asm volatile("global_load_async_to_lds_b32 %0, %1, %2"
                 :
                 : "v"(lds_addr), "v"(goff_bytes), "s"(gbase)
                 : "memory");
}

__device__ __forceinline__ void wait_async0() {
    asm volatile("s_wait_asynccnt 0" ::: "memory");
}

// logaddexp via hardware transcendentals (v_exp_f32 / v_log_f32).
// Correction term is in [0, ln2]; approx error << f32 accumulation rounding.
__device__ __forceinline__ float lae(float x, float y) {
    float mx = fmaxf(x, y);
    float d  = fminf(x, y) - mx;          // <= 0; NEG-NEG -> 0 handled fine
    return mx + __logf(1.0f + __expf(d));
}

// ---------------------------------------------------------------------------
// One workgroup per batch item. 256 threads x 4 consecutive columns each
// (1024 >= TLEN). Per DP step, the 4 in-thread logaddexps are independent
// (all read i-1 state) -> ILP 4 into the trans pipe. Only one neighbor value
// crosses threads per step: in-wave via __shfl_up, across waves via a
// ping-pong LDS slot, so a single s_barrier per frame suffices. Frame rows
// (256 f32 = 1KB) are staged into double-buffered LDS with async copies
// overlapped with the compute.
// ---------------------------------------------------------------------------
__global__ __launch_bounds__(BLOCK) void align_alpha_kernel(
    const float* __restrict__ scores,   // [N, B, V]
    const int*   __restrict__ targets,  // [B, T]
    const int*   __restrict__ in_len,   // [B]
    const int*   __restrict__ tg_len,   // [B]
    float*       __restrict__ partial)  // [B] out: alpha[iL-1, tL-1]
{
    __shared__ float sc[2][VOCAB];      // double-buffered frame rows
    __shared__ float bnd[2][NWAVES];    // ping-pong wave-boundary alphas

    const int b    = blockIdx.x;
    const int tid  = threadIdx.x;
    const int lane = tid & 31;
    const int wv   = tid >> 5;

    const int iL = in_len[b];           // frames to process (1000..2000)
    const int tL = tg_len[b];           // target length      (500..1000)

    // Coalesced int4 load of this thread's 4 token ids (b*TLEN*4 and tid*16
    // are both 16B-aligned; threads >= TLEN/COLS own only junk columns).
    int t0 = 0, t1 = 0, t2 = 0, t3 = 0;
    if (tid < TLEN / COLS) {
        const int4 tv = *(const int4*)(targets + (size_t)b * TLEN + tid * COLS);
        t0 = tv.x & (VOCAB - 1); t1 = tv.y & (VOCAB - 1);
        t2 = tv.z & (VOCAB - 1); t3 = tv.w & (VOCAB - 1);
    }

    // Base points at scores[0, b, 0]; frame i element v is at byte offset
    // (i*BATCH*VOCAB + v) * 4  (max ~65.5 MB, fits in u32).
    const float* row_base = scores + (size_t)b * VOCAB;

    // Prologue: frame 0 row -> sc[0]  (256 threads load 256 dwords)
    async_load_b32(row_base, (unsigned)(tid * 4), lds_off(&sc[0][tid]));
    wait_async0();
    __syncthreads();

    // alpha[0, j] = (j==0) ? emit[0,0] : NEG
    float a0 = (tid == 0) ? sc[0][t0] : NEGV;
    float a1 = NEGV, a2 = NEGV, a3 = NEGV;
    if (lane == 31) bnd[0][wv] = a3;    // column wv*128+127 at i=0 (= NEG)

    int cur = 0, p = 0;

    // Prefetch frame 1 -> sc[1]
    if (iL > 1) {
        unsigned off = (unsigned)((1u * BATCH * VOCAB + (unsigned)tid) * 4u);
        async_load_b32(row_base, off, lds_off(&sc[1][tid]));
    }
    wait_async0();
    __syncthreads();

    for (int i = 1; i < iL; ++i) {
        cur ^= 1;

        // Overlap: kick off frame i+1 into the buffer we are NOT reading.
        if (i + 1 < iL) {
            unsigned off = (unsigned)(((unsigned)(i + 1) * BATCH * VOCAB
                                       + (unsigned)tid) * 4u);
            async_load_b32(row_base, off, lds_off(&sc[cur ^ 1][tid]));
        }

        // Incoming neighbor: old a3 of the left thread (column tid*4 - 1).
        float up   = __shfl_up(a3, 1, 32);
        float bv   = bnd[p][wv ? wv - 1 : 0];
        float left = lane ? up : (wv ? bv : NEGV);

        // Emission gathers from the staged row (LDS broadcast-ish reads).
        float e0 = sc[cur][t0], e1 = sc[cur][t1];
        float e2 = sc[cur][t2], e3 = sc[cur][t3];

        // Four independent recurrences (all consume i-1 state) -> ILP 4.
        float n0 = e0 + lae(left, a0);
        float n1 = e1 + lae(a0, a1);
        float n2 = e2 + lae(a1, a2);
        float n3 = e3 + lae(a2, a3);

        if (lane == 31) bnd[p ^ 1][wv] = n3;   // publish boundary for step i+1
        a0 = n0; a1 = n1; a2 = n2; a3 = n3;
        p ^= 1;

        wait_async0();      // next frame row landed in LDS
        __syncthreads();    // boundary + row visible to all waves
    }

    // alpha[iL-1, tL-1] lives in thread (tL-1)/4, sub-column (tL-1)%4.
    const int jf = tL - 1;
    if ((jf >> 2) == tid) {
        const int c = jf & 3;
        float r = (c == 0) ? a0 : (c == 1) ? a1 : (c == 2) ? a2 : a3;
        partial[b] = r;
    }
}

// Deterministic final reduction: loss = -sum(partial)/BATCH
__global__ void reduce_loss_kernel(const float* __restrict__ partial,
                                   float* __restrict__ out) {
    if (threadIdx.x == 0) {
        float s = 0.0f;
        for (int i = 0; i < BATCH; ++i) s += partial[i];
        out[0] = -s / (float)BATCH;
    }
}

extern "C" void kernel_launch(void* const* d_in, const int* in_sizes, int n_in,
                              void* d_out, int out_size, void* d_ws, size_t ws_size,
                              hipStream_t stream) {
    const float* scores  = (const float*)d_in[0];  // [2000, 32, 256] f32
    const int*   targets = (const int*)  d_in[1];  // [32, 1000] i32
    const int*   ilen    = (const int*)  d_in[2];  // [32] i32
    const int*   tlen    = (const int*)  d_in[3];  // [32] i32
    float*       partial = (float*)d_ws;           // 32 floats scratch

    align_alpha_kernel<<<BATCH, BLOCK, 0, stream>>>(scores, targets, ilen, tlen, partial);
    reduce_loss_kernel<<<1, 32, 0, stream>>>(partial, (float*)d_out);
}